// RelGraphEncoderTG_28286654611821
// MI455X (gfx1250) — compile-verified
//
#include <hip/hip_runtime.h>
#include <hip/hip_bf16.h>

#define NN      65536      // nodes
#define NE      524288     // edges
#define DM      256        // d_model
#define NRELS   32
#define NBASES  8
#define NCOLS   2304       // 8*256 basis outputs + 256 root output
#define LN_EPS  1e-5f

typedef float v2f __attribute__((ext_vector_type(2)));
typedef float v8f __attribute__((ext_vector_type(8)));
typedef unsigned int u32;
typedef unsigned long long u64;
typedef u32 v4u __attribute__((ext_vector_type(4)));
typedef int  v4i __attribute__((ext_vector_type(4)));
typedef int  v8i __attribute__((ext_vector_type(8)));

// ---------------------------------------------------------------- utilities
__global__ void zero_u32_kernel(unsigned* __restrict__ p, int n) {
    int i = blockIdx.x * blockDim.x + threadIdx.x;
    if (i < n) p[i] = 0u;
}

__global__ void zero_f32_kernel(float* __restrict__ p, int n) {
    int i = blockIdx.x * blockDim.x + threadIdx.x;
    if (i < n) p[i] = 0.0f;
}

// h = valid ? x : 0
__global__ void init_h_kernel(const float* __restrict__ x,
                              const int* __restrict__ valid,
                              float* __restrict__ h) {
    int i = blockIdx.x * blockDim.x + threadIdx.x;   // over NN*DM
    int row = i >> 8;
    h[i] = valid[row] ? x[i] : 0.0f;
}

// cnt[dst*NRELS + rel] += 1
__global__ void count_kernel(const int* __restrict__ ei,
                             const int* __restrict__ et,
                             unsigned* __restrict__ cnt) {
    int e = blockIdx.x * blockDim.x + threadIdx.x;
    if (e >= NE) return;
    int dst = ei[NE + e];
    int r   = et[e];
    atomicAdd(&cnt[dst * NRELS + r], 1u);
}

// ---------------------------------------------------------------- WMMA GEMM
// C[NN x NCOLS] = A[NN x 256] * Bcat[256 x NCOLS]
// Bcat cols [0,2048)   -> basisL[b][k][e], b = col>>8, e = col&255
// Bcat cols [2048,2304) -> rootL[k][e],   e = col-2048
// Block tile 64 rows x 128 cols, K chunked by 64 through LDS.
// A chunk staged via TDM (tensor_load_to_lds) with HW padding to stride 68;
// B chunk staged manually (transposed) so B fragments are contiguous b64 loads.
__global__ __launch_bounds__(256) void gemm_kernel(
    const float* __restrict__ A,
    const float* __restrict__ basisL,
    const float* __restrict__ rootL,
    float* __restrict__ C) {

    __shared__ float sA[64 * 68];    // row-major, stride 68 (TDM pad: 64dw + 4dw)
    __shared__ float sB[128 * 68];   // TRANSPOSED: sB[col*68 + k]

    const int tid    = threadIdx.x;
    const int row0   = blockIdx.x * 64;
    const int colBlk = blockIdx.y * 128;

    const float* Bbase = (colBlk < 2048)
        ? (basisL + (size_t)(colBlk >> 8) * 65536 + (colBlk & 255))
        : (rootL + (colBlk - 2048));

    const int lane  = tid & 31;
    const int wave  = tid >> 5;
    const int wr    = (wave & 3) * 16;   // wave row offset in tile
    const int wc    = (wave >> 2) * 64;  // wave col offset in tile
    const int m     = lane & 15;         // A row / B col within fragment
    const int khalf = (lane >> 4) * 2;   // k sub-offset per fragment layout

    // LDS byte offset of sA within this block's LDS allocation
    // (generic pointer to LDS: low 32 bits of the flat address are the LDS offset)
    const u32 ldsA = (u32)(size_t)(void*)sA;

    v8f acc0 = {0,0,0,0,0,0,0,0};
    v8f acc1 = {0,0,0,0,0,0,0,0};
    v8f acc2 = {0,0,0,0,0,0,0,0};
    v8f acc3 = {0,0,0,0,0,0,0,0};

    for (int kc = 0; kc < 256; kc += 64) {
        // --- stage A chunk via Tensor Data Mover: 64 rows x 64 f32, stride 256,
        //     LDS-padded 4 dwords after every 64 dwords -> stride 68.
        if (wave == 0) {
            u64 gaddr = (u64)(const void*)(A + (size_t)row0 * DM + kc);
            v4u g0;
            g0.x = 1u;                                   // count=1, user mode
            g0.y = ldsA;                                 // lds_addr (bytes)
            g0.z = (u32)gaddr;                           // global_addr[31:0]
            g0.w = (u32)(gaddr >> 32) | (2u << 30);      // global_addr[56:32] | type=2
            v8i g1;
            g1[0] = (2 << 16)      // data_size = 4 bytes
                  | (1 << 20)      // pad_enable
                  | (5 << 22)      // pad_interval: 2<<5 = 64 dwords
                  | (3 << 25);     // pad_amount: 3+1 = 4 dwords
            g1[1] = (64 << 16);            // tensor_dim0[15:0]=64
            g1[2] = (64 << 16);            // tensor_dim0 hi=0 | tensor_dim1[15:0]=64
            g1[3] = (64 << 16);            // tensor_dim1 hi=0 | tile_dim0=64
            g1[4] = 64;                    // tile_dim1=64 | tile_dim2=0
            g1[5] = 256;                   // tensor_dim0_stride[31:0]=256
            g1[6] = 0;                     // stride0 hi | stride1 lo
            g1[7] = 0;                     // stride1 hi
            v4i g2 = {0, 0, 0, 0};
            v4i g3 = {0, 0, 0, 0};
            v8i g4 = {0, 0, 0, 0, 0, 0, 0, 0};
            __builtin_amdgcn_tensor_load_to_lds(g0, g1, g2, g3, g4, 0);
        }
        // --- stage B chunk transposed: 64 k x 128 cols (2048 float4, 8/thread)
        #pragma unroll
        for (int s = 0; s < 8; ++s) {
            int i  = s * 256 + tid;
            int k  = i >> 5;
            int cq = (i & 31) * 4;
            float4 v = *(const float4*)(Bbase + (size_t)(kc + k) * DM + cq);
            sB[(cq + 0) * 68 + k] = v.x;
            sB[(cq + 1) * 68 + k] = v.y;
            sB[(cq + 2) * 68 + k] = v.z;
            sB[(cq + 3) * 68 + k] = v.w;
        }
        if (wave == 0) {
            __builtin_amdgcn_s_wait_tensorcnt(0);   // A tile landed in LDS
        }
        __syncthreads();

        #pragma unroll
        for (int k = 0; k < 64; k += 4) {
            v2f a  = *(const v2f*)(&sA[(wr + m) * 68 + k + khalf]);
            v2f b0 = *(const v2f*)(&sB[(wc +  0 + m) * 68 + k + khalf]);
            v2f b1 = *(const v2f*)(&sB[(wc + 16 + m) * 68 + k + khalf]);
            v2f b2 = *(const v2f*)(&sB[(wc + 32 + m) * 68 + k + khalf]);
            v2f b3 = *(const v2f*)(&sB[(wc + 48 + m) * 68 + k + khalf]);
            acc0 = __builtin_amdgcn_wmma_f32_16x16x4_f32(false, a, false, b0, (short)0, acc0, false, false);
            acc1 = __builtin_amdgcn_wmma_f32_16x16x4_f32(false, a, false, b1, (short)0, acc1, false, false);
            acc2 = __builtin_amdgcn_wmma_f32_16x16x4_f32(false, a, false, b2, (short)0, acc2, false, false);
            acc3 = __builtin_amdgcn_wmma_f32_16x16x4_f32(false, a, false, b3, (short)0, acc3, false, false);
        }
        __syncthreads();
    }

    // C/D layout: VGPR v, lanes 0-15 -> (M=v, N=lane); lanes 16-31 -> (M=8+v)
    const int rbase = row0 + wr + (lane >> 4) * 8;
    v8f accs[4] = {acc0, acc1, acc2, acc3};
    #pragma unroll
    for (int t = 0; t < 4; ++t) {
        int col = colBlk + wc + t * 16 + m;
        #pragma unroll
        for (int v = 0; v < 8; ++v) {
            C[(size_t)(rbase + v) * NCOLS + col] = accs[t][v];
        }
    }
}

// ---------------------------------------------------------------- edge pass
// one wave per edge: msg = norm * sum_b comp[rel,b] * xb[src,b,:]; atomic add to agg[dst]
__global__ __launch_bounds__(256) void edge_kernel(
    const float* __restrict__ xb,
    const int* __restrict__ ei,
    const int* __restrict__ et,
    const unsigned* __restrict__ cnt,
    const float* __restrict__ compL,
    float* __restrict__ agg) {

    const int lane = threadIdx.x & 31;
    const int e = blockIdx.x * 8 + (threadIdx.x >> 5);
    if (e >= NE) return;

    const int src = ei[e];
    const int dst = ei[NE + e];
    const int r   = et[e];

    unsigned c = cnt[dst * NRELS + r];
    float norm = 1.0f / (float)(c == 0u ? 1u : c);

    float cb[NBASES];
    #pragma unroll
    for (int b = 0; b < NBASES; ++b) cb[b] = compL[r * NBASES + b];

    const float* __restrict__ xrow = xb + (size_t)src * NCOLS;
    float* __restrict__ arow = agg + (size_t)dst * DM;

    #pragma unroll
    for (int j = 0; j < 2; ++j) {
        int col = j * 128 + lane * 4;
        float4 s = {0.f, 0.f, 0.f, 0.f};
        #pragma unroll
        for (int b = 0; b < NBASES; ++b) {
            float4 v = *(const float4*)(xrow + b * DM + col);
            s.x += cb[b] * v.x;
            s.y += cb[b] * v.y;
            s.z += cb[b] * v.z;
            s.w += cb[b] * v.w;
        }
        atomicAdd(&arow[col + 0], s.x * norm);
        atomicAdd(&arow[col + 1], s.y * norm);
        atomicAdd(&arow[col + 2], s.z * norm);
        atomicAdd(&arow[col + 3], s.w * norm);
    }
}

// ---------------------------------------------------------------- fused epilogue
// out = agg + (h@root, from xb cols 2048..) + bias; gelu(exact); residual; layernorm
__global__ __launch_bounds__(256) void fuse_kernel(
    const float* __restrict__ hin,
    const float* __restrict__ agg,
    const float* __restrict__ xb,
    const float* __restrict__ biasL,
    const float* __restrict__ gammaL,
    const float* __restrict__ betaL,
    const int* __restrict__ valid,
    int apply_mask,
    float* __restrict__ hout) {

    __shared__ float s1[256];
    __shared__ float s2[256];

    const int row = blockIdx.x;
    const int c   = threadIdx.x;

    float h = hin[(size_t)row * DM + c];
    float o = agg[(size_t)row * DM + c]
            + xb[(size_t)row * NCOLS + 2048 + c]
            + biasL[c];
    float g = 0.5f * o * (1.0f + erff(o * 0.7071067811865476f));
    float y = h + g;

    s1[c] = y;
    s2[c] = y * y;
    __syncthreads();
    #pragma unroll
    for (int off = 128; off > 0; off >>= 1) {
        if (c < off) { s1[c] += s1[c + off]; s2[c] += s2[c + off]; }
        __syncthreads();
    }
    float mu  = s1[0] * (1.0f / 256.0f);
    float var = s2[0] * (1.0f / 256.0f) - mu * mu;
    float out = (y - mu) * rsqrtf(var + LN_EPS) * gammaL[c] + betaL[c];
    if (apply_mask && !valid[row]) out = 0.0f;
    hout[(size_t)row * DM + c] = out;
}

// ---------------------------------------------------------------- launcher
extern "C" void kernel_launch(void* const* d_in, const int* in_sizes, int n_in,
                              void* d_out, int out_size, void* d_ws, size_t ws_size,
                              hipStream_t stream) {
    const float* x_flat = (const float*)d_in[0];
    const int*   ei     = (const int*)d_in[1];     // [2, NE]
    const int*   et     = (const int*)d_in[2];     // [NE]
    const int*   valid  = (const int*)d_in[3];     // [NN]
    const float* basis  = (const float*)d_in[4];   // [2, 8, 256, 256]
    const float* comp   = (const float*)d_in[5];   // [2, 32, 8]
    const float* root   = (const float*)d_in[6];   // [2, 256, 256]
    const float* bias   = (const float*)d_in[7];   // [2, 256]
    const float* gamma  = (const float*)d_in[8];   // [2, 256]
    const float* beta   = (const float*)d_in[9];   // [2, 256]
    float* out = (float*)d_out;

    char* ws = (char*)d_ws;
    unsigned* cnt = (unsigned*)(ws);                                 // 8 MB
    float* h0     = (float*)(ws + (size_t)8   * 1024 * 1024);        // 64 MB
    float* agg    = (float*)(ws + (size_t)72  * 1024 * 1024);        // 64 MB
    float* xb     = (float*)(ws + (size_t)136 * 1024 * 1024);        // 576 MB

    // edge-count normalization (layer-independent)
    zero_u32_kernel<<<(NN * NRELS + 255) / 256, 256, 0, stream>>>(cnt, NN * NRELS);
    count_kernel<<<(NE + 255) / 256, 256, 0, stream>>>(ei, et, cnt);

    // h0 = valid ? x : 0
    init_h_kernel<<<(NN * DM + 255) / 256, 256, 0, stream>>>(x_flat, valid, h0);

    for (int l = 0; l < 2; ++l) {
        const float* hin  = (l == 0) ? h0 : out;
        float*       hout = out;   // layer0 -> d_out, layer1 in-place on d_out (row-local)

        zero_f32_kernel<<<(NN * DM + 255) / 256, 256, 0, stream>>>(agg, NN * DM);

        dim3 gg(NN / 64, NCOLS / 128);
        gemm_kernel<<<gg, 256, 0, stream>>>(
            hin,
            basis + (size_t)l * NBASES * DM * DM,
            root  + (size_t)l * DM * DM,
            xb);

        edge_kernel<<<NE / 8, 256, 0, stream>>>(
            xb, ei, et, cnt, comp + (size_t)l * NRELS * NBASES, agg);

        fuse_kernel<<<NN, 256, 0, stream>>>(
            hin, agg, xb,
            bias  + (size_t)l * DM,
            gamma + (size_t)l * DM,
            beta  + (size_t)l * DM,
            valid, (l == 1) ? 1 : 0,
            hout);
    }
}